// ResidualVectorQuantize_34694745817196
// MI455X (gfx1250) — compile-verified
//
#include <hip/hip_runtime.h>
#include <math.h>

// ---------------- problem constants ----------------
#define B_    16
#define D_    512
#define T_    4096
#define NQ    4
#define K_    1024
#define CD    8
#define TILE  64                    // tokens per workgroup tile
#define NBLK  (B_ * (T_ / TILE))    // 1024 main blocks

// output layout (floats): z_q | codes | losses
#define ZQ_ELEMS    (B_ * D_ * T_)          // 33554432
#define CODES_OFF   ZQ_ELEMS
#define CODES_ELEMS (NQ * B_ * T_)          // 262144
#define LOSS_OFF    (CODES_OFF + CODES_ELEMS)

// workspace layout (floats)
#define WS_WIN   0                  // [NQ][CD][D]   16384
#define WS_WOUT  16384              // [NQ][D][CD]   16384
#define WS_CBN   32768              // [NQ][K][CD]   32768
#define WS_PART  65536              // [NBLK] loss partials

typedef float v2f __attribute__((ext_vector_type(2)));
typedef float v8f __attribute__((ext_vector_type(8)));
typedef unsigned int u32;
typedef u32 uint32x4 __attribute__((ext_vector_type(4)));
typedef int  int32x8 __attribute__((ext_vector_type(8)));
typedef int  int32x4 __attribute__((ext_vector_type(4)));

#if defined(__gfx1250__) && __has_builtin(__builtin_amdgcn_tensor_load_to_lds)
#define USE_TDM 1
#else
#define USE_TDM 0
#endif

#if USE_TDM
// Build D# groups per CDNA5 ISA 8.3/8.4 and issue TENSOR_LOAD_TO_LDS (2D tile).
// tensor: [tensor_d1 rows x tensor_d0 cols] of f32, row stride stride0 (elems)
// tile:   [tile_d1 rows x tile_d0 cols] starting at gptr -> LDS lds_addr
__device__ __forceinline__ void tdm_load_2d(u32 lds_addr, const void* gptr,
                                            u32 tensor_d0, u32 tensor_d1,
                                            u32 tile_d0,   u32 tile_d1,
                                            u32 stride0) {
    unsigned long long ga = (unsigned long long)gptr;
    uint32x4 g0;
    g0[0] = 1u;                                        // count=1, user descriptor
    g0[1] = lds_addr;                                  // [63:32] lds_addr
    g0[2] = (u32)(ga & 0xffffffffu);                   // global_addr[31:0]
    g0[3] = (u32)((ga >> 32) & 0x01ffffffu) | (2u << 30); // global_addr[56:32] | type=2
    int32x8 g1;
    g1[0] = (int)(2u << 16);                           // data_size = 4 bytes
    g1[1] = (int)((tensor_d0 & 0xffffu) << 16);        // tensor_dim0[15:0]
    g1[2] = (int)((tensor_d0 >> 16) | ((tensor_d1 & 0xffffu) << 16));
    g1[3] = (int)((tensor_d1 >> 16) | (tile_d0 << 16)); // tile_dim0
    g1[4] = (int)(tile_d1 & 0xffffu);                  // tile_dim1 (tile_dim2=0)
    g1[5] = (int)stride0;                              // tensor_dim0_stride[31:0]
    g1[6] = 0;
    g1[7] = 0;
    int32x4 z4 = {0, 0, 0, 0};
    int32x8 z8 = {0, 0, 0, 0, 0, 0, 0, 0};
    __builtin_amdgcn_tensor_load_to_lds(g0, g1, z4, z4, z8, 0);
}
#endif

// ---------------- precompute: weight-norm weights + normalized codebooks ----
__global__ void rvq_prep(const float* __restrict__ in_v,  const float* __restrict__ in_g,
                         const float* __restrict__ out_v, const float* __restrict__ out_g,
                         const float* __restrict__ cb,    float* __restrict__ ws) {
    int r = blockIdx.x * blockDim.x + threadIdx.x;
    if (r < NQ * CD) {                                   // w_in rows, norm over D
        const float* v = in_v + (size_t)r * D_;
        float s = 0.f;
        for (int d = 0; d < D_; ++d) s += v[d] * v[d];
        float sc = in_g[r] / fmaxf(sqrtf(s), 1e-12f);
        float* w = ws + WS_WIN + (size_t)r * D_;
        for (int d = 0; d < D_; ++d) w[d] = v[d] * sc;
    } else if (r < NQ * CD + NQ * D_) {                  // w_out rows, norm over CD
        int rr = r - NQ * CD;
        const float* v = out_v + (size_t)rr * CD;
        float s = 0.f;
        for (int k = 0; k < CD; ++k) s += v[k] * v[k];
        float sc = out_g[rr] / fmaxf(sqrtf(s), 1e-12f);
        float* w = ws + WS_WOUT + (size_t)rr * CD;
        for (int k = 0; k < CD; ++k) w[k] = v[k] * sc;
    } else if (r < NQ * CD + NQ * D_ + NQ * K_) {        // codebook rows normalized
        int rr = r - NQ * CD - NQ * D_;
        const float* v = cb + (size_t)rr * CD;
        float s = 0.f;
        for (int k = 0; k < CD; ++k) s += v[k] * v[k];
        float sc = 1.0f / fmaxf(sqrtf(s), 1e-12f);
        float* w = ws + WS_CBN + (size_t)rr * CD;
        for (int k = 0; k < CD; ++k) w[k] = v[k] * sc;
    }
}

// ---------------- main fused RVQ kernel ----------------
__global__ __launch_bounds__(256) void rvq_main(
        const float* __restrict__ z,
        const float* __restrict__ in_b,    // [NQ][CD]
        const float* __restrict__ out_b,   // [NQ][D]
        const float* __restrict__ cb,      // [NQ][K][CD] raw codebooks
        const float* __restrict__ ws,
        float* __restrict__ wsp,           // loss partials [NBLK]
        float* __restrict__ out) {
    __shared__ float R[D_ * TILE];        // residual tile [d][t]   128 KB
    __shared__ float Wbuf[CD * D_];       // staged w_in / w_out     16 KB
    __shared__ float ZEpart[2 * CD * TILE]; // z_e k-half partials    4 KB
    __shared__ float ZE[CD * TILE];       // z_e  [k][t]
    __shared__ float EN[CD * TILE];       // e_n  [k][t]
    __shared__ float ZR[CD * TILE];       // z_rot[k][t]
    __shared__ float NEn[TILE];           // ||e|| per token
    __shared__ float candv[2 * TILE];
    __shared__ int   candi[2 * TILE];
    __shared__ int   IDXs[TILE];
    __shared__ float lossb[TILE];

    const int tid  = threadIdx.x;
    const int wave = tid >> 5;
    const int lane = tid & 31;
    const int blk  = blockIdx.x;
    const int b    = blk >> 6;              // 64 tiles per batch
    const int t0   = (blk & 63) * TILE;
    const float* zb = z + (size_t)b * D_ * T_;

    const float* w_in_all  = ws + WS_WIN;
    const float* w_out_all = ws + WS_WOUT;
    const float* cbn_all   = ws + WS_CBN;

    float loss_acc = 0.f;  // meaningful on tid 0

    // ---- residual tile -> LDS ----
#if USE_TDM
    if (wave == 0) {
        tdm_load_2d((u32)(uintptr_t)R, zb + t0, T_, D_, TILE, D_, T_);
        __builtin_amdgcn_s_wait_tensorcnt(0);
    }
#else
    for (int i = tid; i < D_ * TILE / 4; i += 256) {
        int d  = i >> 4;
        int t4 = (i & 15) << 2;
        float4 v = *(const float4*)(zb + (size_t)d * T_ + t0 + t4);
        *(float4*)(R + d * TILE + t4) = v;
    }
#endif
    __syncthreads();

    for (int s = 0; s < NQ; ++s) {
        // ---- stage w_in [CD][D] into LDS ----
#if USE_TDM
        if (wave == 0) {
            tdm_load_2d((u32)(uintptr_t)Wbuf, w_in_all + (size_t)s * CD * D_,
                        CD * D_, 1, CD * D_, 1, CD * D_);
            __builtin_amdgcn_s_wait_tensorcnt(0);
        }
#else
        for (int i = tid; i < CD * D_ / 4; i += 256)
            *(float4*)(Wbuf + i * 4) = *(const float4*)(w_in_all + (size_t)s * CD * D_ + i * 4);
#endif
        __syncthreads();

        // ---- z_e = w_in @ R via WMMA f32 16x16x4 (M=16 pad of 8 channels) ----
        {
            int ntile = wave & 3;                // 16-token slab
            int khalf = wave >> 2;               // K 0..255 / 256..511
            int m   = lane & 15;
            bool mok = (m < 8);
            int klo = (lane < 16) ? 0 : 2;
            int t   = ntile * 16 + m;
            const float* wrow = Wbuf + (mok ? m : 0) * D_;
            const float* Rt   = R + t;
            v8f acc = {};
            int kbase = khalf * 256 + klo;
            for (int ks = 0; ks < 64; ++ks) {
                int k = kbase + ks * 4;
                v2f Aw = *(const v2f*)(wrow + k);
                v2f A;
                A.x = mok ? Aw.x : 0.f;
                A.y = mok ? Aw.y : 0.f;
                v2f Bv;
                Bv.x = Rt[(k + 0) * TILE];
                Bv.y = Rt[(k + 1) * TILE];
                acc = __builtin_amdgcn_wmma_f32_16x16x4_f32(false, A, false, Bv, (short)0, acc, false, false);
            }
            if (lane < 16) {                     // rows m = 0..7 live in lanes 0..15
                #pragma unroll
                for (int r = 0; r < 8; ++r)
                    ZEpart[(khalf * 8 + r) * TILE + t] = acc[r];
            }
        }
        __syncthreads();

        // ---- combine k-halves + bias, L2 normalize per token ----
        if (tid < TILE) {
            float e[CD]; float ss = 0.f;
            #pragma unroll
            for (int k = 0; k < CD; ++k) {
                float v = ZEpart[k * TILE + tid] + ZEpart[(8 + k) * TILE + tid] + in_b[s * CD + k];
                e[k] = v; ss += v * v;
                ZE[k * TILE + tid] = v;
            }
            float nrm = sqrtf(ss);
            NEn[tid] = nrm;
            float inv = 1.0f / fmaxf(nrm, 1e-12f);
            #pragma unroll
            for (int k = 0; k < CD; ++k) EN[k * TILE + tid] = e[k] * inv;
        }
        __syncthreads();

        // ---- code search via WMMA (K=8 as 2 x K=4) ----
        {
            int mtile = wave & 3;
            int nhalf = wave >> 2;
            int m   = mtile * 16 + (lane & 15);
            int khi = (lane < 16) ? 0 : 2;
            v2f A0, A1;
            A0.x = EN[(khi + 0) * TILE + m]; A0.y = EN[(khi + 1) * TILE + m];
            A1.x = EN[(khi + 4) * TILE + m]; A1.y = EN[(khi + 5) * TILE + m];

            float bestv[8]; int besti[8];
            #pragma unroll
            for (int r = 0; r < 8; ++r) { bestv[r] = -3.0e38f; besti[r] = 0; }

            const float* cbn = cbn_all + (size_t)s * K_ * CD;
            for (int nt = 0; nt < 32; ++nt) {
                int n = (nhalf * 32 + nt) * 16 + (lane & 15);
                const float* cr = cbn + (size_t)n * CD + khi;
                v2f B0 = { cr[0], cr[1] };
                v2f B1 = { cr[4], cr[5] };
                v8f c8 = {};
                c8 = __builtin_amdgcn_wmma_f32_16x16x4_f32(false, A0, false, B0, (short)0, c8, false, false);
                c8 = __builtin_amdgcn_wmma_f32_16x16x4_f32(false, A1, false, B1, (short)0, c8, false, false);
                #pragma unroll
                for (int r = 0; r < 8; ++r) {
                    float v = c8[r];
                    bool take = v > bestv[r];
                    bestv[r] = take ? v : bestv[r];
                    besti[r] = take ? n : besti[r];
                }
            }
            // branchless reduce across the 16 lanes of each half
            #pragma unroll
            for (int off = 8; off >= 1; off >>= 1) {
                #pragma unroll
                for (int r = 0; r < 8; ++r) {
                    float ov = __shfl_xor(bestv[r], off, 32);
                    int   oi = __shfl_xor(besti[r], off, 32);
                    bool take = (ov > bestv[r]) | ((ov == bestv[r]) & (oi < besti[r]));
                    bestv[r] = take ? ov : bestv[r];
                    besti[r] = take ? oi : besti[r];
                }
            }
            if ((lane & 15) == 0) {
                int half = lane >> 4;
                #pragma unroll
                for (int r = 0; r < 8; ++r) {
                    int tok = mtile * 16 + half * 8 + r;
                    candv[nhalf * TILE + tok] = bestv[r];
                    candi[nhalf * TILE + tok] = besti[r];
                }
            }
        }
        __syncthreads();

        // ---- combine the two code halves, emit codes ----
        if (tid < TILE) {
            float v0 = candv[tid], v1 = candv[TILE + tid];
            int   i0 = candi[tid], i1 = candi[TILE + tid];
            int idx = (v1 > v0 || (v1 == v0 && i1 < i0)) ? i1 : i0;
            IDXs[tid] = idx;
            out[CODES_OFF + ((size_t)s * B_ + b) * T_ + t0 + tid] = (float)idx;
        }
        __syncthreads();

        // ---- rotation trick + per-token loss ----
        if (tid < TILE) {
            int idx = IDXs[tid];
            const float* q = cb + ((size_t)s * K_ + idx) * CD;
            float e[CD], en[CD], qv[CD];
            float qs = 0.f, lacc = 0.f;
            #pragma unroll
            for (int k = 0; k < CD; ++k) {
                e[k]  = ZE[k * TILE + tid];
                en[k] = EN[k * TILE + tid];
                qv[k] = q[k];
                qs += qv[k] * qv[k];
                float dlt = e[k] - qv[k];
                lacc += dlt * dlt;
            }
            float qn  = sqrtf(qs);
            float qiv = 1.0f / fmaxf(qn, 1e-12f);
            float rv[CD]; float rs = 0.f;
            #pragma unroll
            for (int k = 0; k < CD; ++k) { float t2 = en[k] + qv[k] * qiv; rv[k] = t2; rs += t2 * t2; }
            float riv = 1.0f / fmaxf(sqrtf(rs), 1e-12f);
            float re = 0.f, ene = 0.f;
            #pragma unroll
            for (int k = 0; k < CD; ++k) { rv[k] *= riv; re += rv[k] * e[k]; ene += en[k] * e[k]; }
            float scale = qn / fmaxf(NEn[tid], 1e-8f);
            #pragma unroll
            for (int k = 0; k < CD; ++k)
                ZR[k * TILE + tid] = scale * (e[k] - 2.f * rv[k] * re + 2.f * (qv[k] * qiv) * ene);
            lossb[tid] = lacc;
        }
        __syncthreads();

        // ---- deterministic per-block loss sum; stage w_out [D][CD] ----
        if (tid == 0) {
            float a = 0.f;
            for (int t = 0; t < TILE; ++t) a += lossb[t];
            loss_acc += a;
        }
#if USE_TDM
        if (wave == 0) {
            tdm_load_2d((u32)(uintptr_t)Wbuf, w_out_all + (size_t)s * D_ * CD,
                        D_ * CD, 1, D_ * CD, 1, D_ * CD);
            __builtin_amdgcn_s_wait_tensorcnt(0);
        }
#else
        for (int i = tid; i < D_ * CD / 4; i += 256)
            *(float4*)(Wbuf + i * 4) = *(const float4*)(w_out_all + (size_t)s * D_ * CD + i * 4);
#endif
        __syncthreads();

        // ---- out-proj via WMMA; subtract from residual in LDS ----
        for (int ti = wave; ti < 128; ti += 8) {
            int mt  = ti >> 2;          // 0..31 : 16-row slab of D
            int ntl = ti & 3;           // 0..3  : 16-token slab
            int d0  = mt * 16;
            int tt0 = ntl * 16;
            int dA  = d0 + (lane & 15);
            int khi = (lane < 16) ? 0 : 2;
            v2f A0 = { Wbuf[dA * CD + khi],     Wbuf[dA * CD + khi + 1] };
            v2f A1 = { Wbuf[dA * CD + khi + 4], Wbuf[dA * CD + khi + 5] };
            int tB = tt0 + (lane & 15);
            v2f B0 = { ZR[(khi + 0) * TILE + tB], ZR[(khi + 1) * TILE + tB] };
            v2f B1 = { ZR[(khi + 4) * TILE + tB], ZR[(khi + 5) * TILE + tB] };
            v8f c8 = {};
            c8 = __builtin_amdgcn_wmma_f32_16x16x4_f32(false, A0, false, B0, (short)0, c8, false, false);
            c8 = __builtin_amdgcn_wmma_f32_16x16x4_f32(false, A1, false, B1, (short)0, c8, false, false);
            int half = lane >> 4;
            #pragma unroll
            for (int r = 0; r < 8; ++r) {
                int d = d0 + half * 8 + r;
                int t = tt0 + (lane & 15);
                float val = c8[r] + out_b[s * D_ + d];
                R[d * TILE + t] -= val;
            }
        }
        __syncthreads();
    }

    // ---- z_q = z - residual_final (z re-read, L2-resident) ----
    for (int i = tid; i < D_ * TILE / 4; i += 256) {
        int d  = i >> 4;
        int t4 = (i & 15) << 2;
        float4 zv = *(const float4*)(zb + (size_t)d * T_ + t0 + t4);
        float4 rv = *(const float4*)(R + d * TILE + t4);
        float4 o;
        o.x = zv.x - rv.x; o.y = zv.y - rv.y; o.z = zv.z - rv.z; o.w = zv.w - rv.w;
        *(float4*)(out + ((size_t)b * D_ + d) * T_ + t0 + t4) = o;
    }
    if (tid == 0) wsp[blk] = loss_acc;   // deterministic partial
}

// ---------------- deterministic loss reduction ----------------
__global__ void rvq_reduce(const float* __restrict__ ws, float* __restrict__ out) {
    __shared__ float sm[256];
    int tid = threadIdx.x;
    const float* p = ws + WS_PART;
    float a = p[tid] + p[tid + 256] + p[tid + 512] + p[tid + 768];
    sm[tid] = a;
    __syncthreads();
    for (int off = 128; off > 0; off >>= 1) {
        if (tid < off) sm[tid] += sm[tid + off];
        __syncthreads();
    }
    if (tid == 0) {
        float v = sm[0] * (1.0f / (float)(B_ * CD * T_));
        out[LOSS_OFF + 0] = v;   // commitment_loss
        out[LOSS_OFF + 1] = v;   // codebook_loss (numerically identical)
    }
}

// ---------------- launch ----------------
extern "C" void kernel_launch(void* const* d_in, const int* in_sizes, int n_in,
                              void* d_out, int out_size, void* d_ws, size_t ws_size,
                              hipStream_t stream) {
    (void)in_sizes; (void)n_in; (void)out_size; (void)ws_size;
    const float* z     = (const float*)d_in[0];
    const float* in_v  = (const float*)d_in[1];
    const float* in_g  = (const float*)d_in[2];
    const float* in_b  = (const float*)d_in[3];
    const float* out_v = (const float*)d_in[4];
    const float* out_g = (const float*)d_in[5];
    const float* out_b = (const float*)d_in[6];
    const float* cb    = (const float*)d_in[7];
    float* out = (float*)d_out;
    float* ws  = (float*)d_ws;

    int prep_rows = NQ * CD + NQ * D_ + NQ * K_;           // 6176
    hipLaunchKernelGGL(rvq_prep, dim3((prep_rows + 255) / 256), dim3(256), 0, stream,
                       in_v, in_g, out_v, out_g, cb, ws);
    hipLaunchKernelGGL(rvq_main, dim3(NBLK), dim3(256), 0, stream,
                       z, in_b, out_b, cb, ws, ws + WS_PART, out);
    hipLaunchKernelGGL(rvq_reduce, dim3(1), dim3(256), 0, stream, ws, out);
}